// SpikeMemoryNetwork_56736517980952
// MI455X (gfx1250) — compile-verified
//
#include <hip/hip_runtime.h>

// Dims (match reference)
#define BB   64
#define TT   200
#define NIN  700
#define NS   50
#define NZ   100
#define NH   512
#define NHO  100
#define NOUT 20

typedef __attribute__((ext_vector_type(2))) float v2f;
typedef __attribute__((ext_vector_type(8))) float v8f;

// LIF: refrac decrement, leak (REST=0, DECAY=0.95), gated injection,
// spike at 0.5, hard reset, refrac=2.  Matches reference order exactly.
__device__ __forceinline__ float lif_step(float& v, float& r, float x) {
  r = fmaxf(r - 1.0f, 0.0f);
  float inj = (r <= 0.0f) ? x : 0.0f;
  v = 0.95f * v + inj;
  float spk = (v >= 0.5f) ? 1.0f : 0.0f;
  if (spk > 0.0f) { v = 0.0f; r = 2.0f; }
  return spk;
}

// 16x64 per-wave WMMA tile (4 accumulators, A fragment reused 4x), K=NH.
// Fragment layouts per ISA 7.12.2 (wave32): A 16x4 = K pair per lane-half;
// B/C/D row-striped across lanes, C/D rows v and v+8 per VGPR.
__device__ __forceinline__ void wmma_16x64(
    const float* __restrict__ A, const float* __restrict__ W,
    int m0, int n0, int lm, int half, v8f acc[4])
{
  for (int k = 0; k < NH; k += 4) {
    const int kk = k + 2 * half;
    v2f a = *(const v2f*)(A + (m0 + lm) * NH + kk);
    const float* w0 = W + (size_t)kk * NH;
    const float* w1 = W + (size_t)(kk + 1) * NH;
#pragma unroll
    for (int c = 0; c < 4; ++c) {
      v2f b;
      b.x = w0[n0 + 16 * c + lm];
      b.y = w1[n0 + 16 * c + lm];
      acc[c] = __builtin_amdgcn_wmma_f32_16x16x4_f32(
          false, a, false, b, (short)0, acc[c], false, false);
    }
  }
}

// ---------------------------------------------------------------------------
// Kernel A: blockIdx 0..3  -> u1 = h_prev @ W_h  (32 waves, 16x64 tiles)
//           blockIdx 4     -> tiny front: x@W_i -> LIF s -> @W_z -> LIF z
// ---------------------------------------------------------------------------
__global__ __launch_bounds__(256) void kA(
    const float* __restrict__ h_prev, const float* __restrict__ W_h,
    float* __restrict__ u1,
    const float* __restrict__ x_t, const float* __restrict__ W_i,
    const float* __restrict__ W_z,
    float* vs, float* rs, float* vz, float* rz,
    float* __restrict__ zspk)
{
  __shared__ float s_spk[BB * NS];   // 12.8 KB
  if (blockIdx.x < 4) {
    const int wave = threadIdx.x >> 5;
    const int lane = threadIdx.x & 31;
    const int half = lane >> 4, lm = lane & 15;
    const int wtile = blockIdx.x * 8 + wave;     // 0..31
    const int m0 = (wtile >> 3) * 16;            // 4 batch tiles
    const int n0 = (wtile & 7) * 64;             // 8 N tiles of 64
    v8f acc[4] = {};
    wmma_16x64(h_prev, W_h, m0, n0, lm, half, acc);
#pragma unroll
    for (int c = 0; c < 4; ++c)
#pragma unroll
      for (int v = 0; v < 8; ++v) {
        const int row = m0 + v + 8 * half;
        u1[row * NH + n0 + 16 * c + lm] = acc[c][v];
      }
  } else {
    for (int idx = threadIdx.x; idx < BB * NS; idx += blockDim.x) {
      const int b = idx / NS, j = idx % NS;
      const float* xr = x_t + b * NIN;
      float d = 0.f;
      for (int k = 0; k < NIN; ++k) d = fmaf(xr[k], W_i[k * NS + j], d);
      float v = vs[idx], r = rs[idx];
      float spk = lif_step(v, r, d);
      vs[idx] = v; rs[idx] = r;
      s_spk[idx] = spk;
    }
    __syncthreads();
    for (int idx = threadIdx.x; idx < BB * NZ; idx += blockDim.x) {
      const int b = idx / NZ, j = idx % NZ;
      float d = 0.f;
      for (int k = 0; k < NS; ++k) d = fmaf(s_spk[b * NS + k], W_z[k * NZ + j], d);
      float v = vz[idx], r = rz[idx];
      float spk = lif_step(v, r, d);
      vz[idx] = v; rz[idx] = r;
      zspk[idx] = spk;
    }
  }
}

// ---------------------------------------------------------------------------
// Kernel B: grid = 64 samples x 8 row-blocks (64 rows each), 256 threads.
//  - recompute cz row for sample b from z spikes (K=100, trivial, redundant)
//  - hidden LIF (ping-pong vh/rh state; only blk 0 writes state/cz/hspk)
//  - fused fast-weight RMW (float4):  a = lam*a + eta*h_i*h_j
//    and recall partials  sum_i h_i * a_old_ij  (combined via LDS, no atomics)
// ---------------------------------------------------------------------------
__global__ __launch_bounds__(256) void kB(
    const float* __restrict__ zspk, const float* __restrict__ W_c,
    const float* __restrict__ u1,
    const float* __restrict__ vh_in, const float* __restrict__ rh_in,
    float* __restrict__ vh_out, float* __restrict__ rh_out,
    float* __restrict__ hspk, float* __restrict__ czg,
    float* __restrict__ Afw, float* __restrict__ part,
    float* __restrict__ nspk,
    const float* __restrict__ lam_p, const float* __restrict__ eta_p)
{
  const int b   = blockIdx.x >> 3;
  const int blk = blockIdx.x & 7;
  __shared__ float zrow[NZ];
  __shared__ float hl[NH];
  __shared__ float psum[NH];

  for (int j = threadIdx.x; j < NZ; j += blockDim.x)
    zrow[j] = zspk[b * NZ + j];
  __syncthreads();

  for (int j = threadIdx.x; j < NH; j += blockDim.x) {
    float d = 0.f;
    for (int k = 0; k < NZ; ++k) d = fmaf(zrow[k], W_c[k * NH + j], d);
    const int idx = b * NH + j;
    float v = vh_in[idx], r = rh_in[idx];
    float spk = lif_step(v, r, d + u1[idx]);
    hl[j] = spk;
    if (blk == 0) {
      vh_out[idx] = v; rh_out[idx] = r;
      hspk[idx] = spk; czg[idx] = d;
    }
  }
  __syncthreads();
  if (blk == 0 && threadIdx.x == 0) {   // h.h for binary spikes
    float s = 0.f;
    for (int j = 0; j < NH; ++j) s += hl[j];
    nspk[b] = s;
  }

  const float lam = lam_p[0], eta = eta_p[0];
  const int r4 = threadIdx.x >> 7;            // row-group 0/1
  const int c  = (threadIdx.x & 127) * 4;     // column (float4)
  const float4 hj = make_float4(hl[c], hl[c + 1], hl[c + 2], hl[c + 3]);
  float4 acc = make_float4(0.f, 0.f, 0.f, 0.f);
  const int i0 = blk * 64;
  for (int i = i0 + r4; i < i0 + 64; i += 2) {
    float4* ap = (float4*)(Afw + ((size_t)b * NH + i) * NH + c);
    __builtin_prefetch((const void*)(Afw + ((size_t)b * NH + i + 4) * NH + c), 0, 1);
    const float hi = hl[i];
    float4 av = *ap;
    acc.x = fmaf(hi, av.x, acc.x);            // recall uses a_old
    acc.y = fmaf(hi, av.y, acc.y);
    acc.z = fmaf(hi, av.z, acc.z);
    acc.w = fmaf(hi, av.w, acc.w);
    const float e = eta * hi;                 // decay + outer-product update
    av.x = fmaf(lam, av.x, e * hj.x);
    av.y = fmaf(lam, av.y, e * hj.y);
    av.z = fmaf(lam, av.z, e * hj.z);
    av.w = fmaf(lam, av.w, e * hj.w);
    *ap = av;
  }
  if (r4 == 0) *(float4*)&psum[c] = acc;
  __syncthreads();
  if (r4 == 1) {
    float4 p = *(const float4*)&psum[c];
    float* pr = part + (size_t)(b * 8 + blk) * NH + c;
    pr[0] = acc.x + p.x; pr[1] = acc.y + p.y;
    pr[2] = acc.z + p.z; pr[3] = acc.w + p.w;
  }
}

// ---------------------------------------------------------------------------
// Kernel C: h_new = hspk@W_h (WMMA 16x64/wave) + cz + lam*sum(partials)
//           + eta*nspk*hspk
// ---------------------------------------------------------------------------
__global__ __launch_bounds__(256) void kC(
    const float* __restrict__ hspk, const float* __restrict__ W_h,
    const float* __restrict__ cz, const float* __restrict__ part,
    const float* __restrict__ nspk,
    const float* __restrict__ lam_p, const float* __restrict__ eta_p,
    float* __restrict__ h_new)
{
  const int wave = threadIdx.x >> 5;
  const int lane = threadIdx.x & 31;
  const int half = lane >> 4, lm = lane & 15;
  const int wtile = blockIdx.x * 8 + wave;
  const int m0 = (wtile >> 3) * 16;
  const int n0 = (wtile & 7) * 64;
  v8f acc[4] = {};
  wmma_16x64(hspk, W_h, m0, n0, lm, half, acc);

  const float lam = lam_p[0], eta = eta_p[0];
#pragma unroll
  for (int c = 0; c < 4; ++c)
#pragma unroll
    for (int v = 0; v < 8; ++v) {
      const int row = m0 + v + 8 * half;
      const int col = n0 + 16 * c + lm;
      float rec = 0.f;
#pragma unroll
      for (int s = 0; s < 8; ++s)
        rec += part[(size_t)(row * 8 + s) * NH + col];
      const float hv = hspk[row * NH + col];
      h_new[row * NH + col] =
          acc[c][v] + cz[row * NH + col] + lam * rec + eta * nspk[row] * hv;
    }
}

// ---------------------------------------------------------------------------
// Kernel D: output layers  h_new@W_ho -> LIF -> @W_o -> LIF -> y_t
// ---------------------------------------------------------------------------
__global__ __launch_bounds__(1024) void kD(
    const float* __restrict__ h, const float* __restrict__ W_ho,
    const float* __restrict__ W_o,
    float* vho, float* rho, float* vo, float* ro,
    float* __restrict__ y_t)
{
  __shared__ float o_spk[BB * NHO];  // 25.6 KB
  for (int idx = threadIdx.x; idx < BB * NHO; idx += blockDim.x) {
    const int b = idx / NHO, j = idx % NHO;
    const float* hr = h + b * NH;
    float d = 0.f;
    for (int k = 0; k < NH; ++k) d = fmaf(hr[k], W_ho[k * NHO + j], d);
    float v = vho[idx], r = rho[idx];
    float spk = lif_step(v, r, d);
    vho[idx] = v; rho[idx] = r;
    o_spk[idx] = spk;
  }
  __syncthreads();
  for (int idx = threadIdx.x; idx < BB * NOUT; idx += blockDim.x) {
    const int b = idx / NOUT, j = idx % NOUT;
    float d = 0.f;
    for (int k = 0; k < NHO; ++k) d = fmaf(o_spk[b * NHO + k], W_o[k * NOUT + j], d);
    float v = vo[idx], r = ro[idx];
    float spk = lif_step(v, r, d);
    vo[idx] = v; ro[idx] = r;
    y_t[idx] = spk;
  }
}

// ---------------------------------------------------------------------------
extern "C" void kernel_launch(void* const* d_in, const int* in_sizes, int n_in,
                              void* d_out, int out_size, void* d_ws, size_t ws_size,
                              hipStream_t stream)
{
  (void)in_sizes; (void)n_in; (void)out_size; (void)ws_size;
  const float* x_in = (const float*)d_in[0];
  const float* W_i  = (const float*)d_in[1];
  const float* W_z  = (const float*)d_in[2];
  const float* W_c  = (const float*)d_in[3];
  const float* W_h  = (const float*)d_in[4];
  const float* W_ho = (const float*)d_in[5];
  const float* W_o  = (const float*)d_in[6];
  const float* lam  = (const float*)d_in[7];
  const float* eta  = (const float*)d_in[8];
  float* out = (float*)d_out;
  float* ws  = (float*)d_ws;

  // Workspace layout: zero-initialized region first (state + fast weights)
  size_t off = 0;
  float* Afw = ws;            off += (size_t)BB * NH * NH;  // 64 MB, L2-resident
  float* vh0 = ws + off;      off += BB * NH;
  float* rh0 = ws + off;      off += BB * NH;
  float* vh1 = ws + off;      off += BB * NH;
  float* rh1 = ws + off;      off += BB * NH;
  float* vs  = ws + off;      off += BB * NS;
  float* rs  = ws + off;      off += BB * NS;
  float* vz  = ws + off;      off += BB * NZ;
  float* rz  = ws + off;      off += BB * NZ;
  float* vho = ws + off;      off += BB * NHO;
  float* rho = ws + off;      off += BB * NHO;
  float* vo  = ws + off;      off += BB * NOUT;
  float* ro  = ws + off;      off += BB * NOUT;
  float* h   = ws + off;      off += BB * NH;               // h state (h_new)
  const size_t zeroBytes = off * sizeof(float);
  float* zspk = ws + off;     off += BB * NZ;
  float* hspk = ws + off;     off += BB * NH;
  float* cz   = ws + off;     off += BB * NH;
  float* u1   = ws + off;     off += BB * NH;
  float* part = ws + off;     off += (size_t)BB * 8 * NH;
  float* nspk = ws + off;     off += BB;

  // Re-zero all mutable state inside the captured graph -> deterministic replay
  hipMemsetAsync(d_ws, 0, zeroBytes, stream);

  for (int t = 0; t < TT; ++t) {
    const float* x_t = x_in + (size_t)t * BB * NIN;
    float* vin  = (t & 1) ? vh1 : vh0;
    float* rin  = (t & 1) ? rh1 : rh0;
    float* vout = (t & 1) ? vh0 : vh1;
    float* rout = (t & 1) ? rh0 : rh1;
    kA<<<5,      256,  0, stream>>>(h, W_h, u1, x_t, W_i, W_z,
                                    vs, rs, vz, rz, zspk);
    kB<<<BB * 8, 256,  0, stream>>>(zspk, W_c, u1, vin, rin, vout, rout,
                                    hspk, cz, Afw, part, nspk, lam, eta);
    kC<<<4,      256,  0, stream>>>(hspk, W_h, cz, part, nspk, lam, eta, h);
    kD<<<1,      1024, 0, stream>>>(h, W_ho, W_o, vho, rho, vo, ro,
                                    out + (size_t)t * BB * NOUT);
  }
}